// GraphConvNetwork_37572373905750
// MI455X (gfx1250) — compile-verified
//
#include <hip/hip_runtime.h>

typedef __attribute__((ext_vector_type(2))) float v2f;
typedef __attribute__((ext_vector_type(8))) float v8f;

#define N_IN   128
#define N_HID  8
#define NPAD   16   // padded N for the 16x16 WMMA tile

// ---------------------------------------------------------------- zero scratch
__global__ __launch_bounds__(256) void k_zero(float* __restrict__ p, long long n) {
  long long i = (long long)blockIdx.x * blockDim.x + threadIdx.x;
  if (i < n) p[i] = 0.0f;
}

// ---------------------------------------------------------------- degrees (float atomics, counts < 2^24 so exact)
__global__ __launch_bounds__(256) void k_degree(const int* __restrict__ src,
                                                const int* __restrict__ dst,
                                                float* __restrict__ deg_out,
                                                float* __restrict__ deg_in,
                                                int n_edges) {
  int i = blockIdx.x * blockDim.x + threadIdx.x;
  if (i >= n_edges) return;
  atomicAdd(&deg_out[src[i]], 1.0f);
  atomicAdd(&deg_in[dst[i]], 1.0f);
}

// ---------------------------------------------------------------- deg -> clip(deg,1)^-0.5 in place
__global__ __launch_bounds__(256) void k_norm(float* __restrict__ d, int n) {
  int i = blockIdx.x * blockDim.x + threadIdx.x;
  if (i >= n) return;
  float v = d[i];
  v = fmaxf(v, 1.0f);
  d[i] = rsqrtf(v);
}

// ---------------------------------------------------------------- layer-1 GEMM via v_wmma_f32_16x16x4_f32
// h1[16 nodes, 8] = (x[16 nodes,128] * norm_out) @ W1[128,8]  (W1 zero-padded to 16 cols)
// One wave handles one 16-node tile; K=128 consumed 4 at a time (32 WMMAs).
__global__ __launch_bounds__(256) void k_gemm1_wmma(const float* __restrict__ x,
                                                    const float* __restrict__ W1,
                                                    const float* __restrict__ norm_out,
                                                    float* __restrict__ h1,
                                                    int n_nodes) {
  __shared__ float Wpad[N_IN * NPAD];   // 8 KB of the 320KB/WGP LDS
  int tid = threadIdx.x;
  for (int i = tid; i < N_IN * NPAD; i += 256) {
    int row = i >> 4, col = i & 15;
    Wpad[i] = (col < N_HID) ? W1[row * N_HID + col] : 0.0f;
  }
  __syncthreads();

  int wave = tid >> 5;
  int lane = tid & 31;
  int half = lane >> 4;   // 0: lanes 0-15, 1: lanes 16-31
  int rc   = lane & 15;   // A-row index / B-col index for this lane

  int node_base = blockIdx.x * 128 + wave * 16;
  if (node_base + 16 > n_nodes) return;   // wave-uniform: EXEC stays all-1s for WMMA

  int arow = node_base + rc;
  float no = norm_out[arow];
  // ISA A-layout (32-bit 16x4): lanes 0-15 hold K=k+0,k+1; lanes 16-31 hold K=k+2,k+3
  const float* xrow = x + (size_t)arow * N_IN + 2 * half;

  v8f acc = {};
  #pragma unroll
  for (int k = 0; k < N_IN; k += 4) {
    v2f a;
    a.x = xrow[k]     * no;
    a.y = xrow[k + 1] * no;
    v2f b;  // B-layout (4x16): rows K=0,1 in lanes 0-15 of vgpr0,1; rows K=2,3 in lanes 16-31
    b.x = Wpad[(k + 2 * half)     * NPAD + rc];
    b.y = Wpad[(k + 2 * half + 1) * NPAD + rc];
    // 8 args: (neg_a, A, neg_b, B, c_mod, C, reuse_a, reuse_b)
    acc = __builtin_amdgcn_wmma_f32_16x16x4_f32(false, a, false, b, (short)0, acc,
                                                false, false);
  }

  // C/D layout: vgpr r, lanes 0-15 -> M=r, lanes 16-31 -> M=r+8; N = rc
  if (rc < N_HID) {
    #pragma unroll
    for (int r = 0; r < 8; ++r) {
      int m = node_base + r + 8 * half;
      h1[(size_t)m * N_HID + rc] = acc[r];
    }
  }
}

// ---------------------------------------------------------------- layer-1 message scatter-add (thread per edge*channel)
__global__ __launch_bounds__(256) void k_scatter1(const int* __restrict__ src,
                                                  const int* __restrict__ dst,
                                                  const float* __restrict__ h1,
                                                  float* __restrict__ agg1,
                                                  int n_edges) {
  long long i = (long long)blockIdx.x * blockDim.x + threadIdx.x;
  long long total = (long long)n_edges * N_HID;
  if (i >= total) return;
  int e = (int)(i >> 3);
  int c = (int)(i & 7);
  int s = src[e];
  int d = dst[e];
  float v = h1[(size_t)s * N_HID + c];
  atomicAdd(&agg1[(size_t)d * N_HID + c], v);
}

// ---------------------------------------------------------------- finish layer 1 + pre-scale layer 2
// h2pre[n] = ( relu(agg1[n,:]*norm_in[n] + b1) . W2 ) * norm_out[n]
__global__ __launch_bounds__(256) void k_node_mid(const float* __restrict__ agg1,
                                                  const float* __restrict__ norm_in,
                                                  const float* __restrict__ norm_out,
                                                  const float* __restrict__ b1,
                                                  const float* __restrict__ W2,
                                                  float* __restrict__ h2pre,
                                                  int n_nodes) {
  int i = blockIdx.x * blockDim.x + threadIdx.x;
  if (i >= n_nodes) return;
  float ni = norm_in[i];
  float s = 0.0f;
  #pragma unroll
  for (int c = 0; c < N_HID; ++c) {
    float h = fmaf(agg1[(size_t)i * N_HID + c], ni, b1[c]);
    h = fmaxf(h, 0.0f);
    s = fmaf(h, W2[c], s);
  }
  h2pre[i] = s * norm_out[i];
}

// ---------------------------------------------------------------- layer-2 scatter-add (thread per edge)
__global__ __launch_bounds__(256) void k_scatter2(const int* __restrict__ src,
                                                  const int* __restrict__ dst,
                                                  const float* __restrict__ h2pre,
                                                  float* __restrict__ acc2,
                                                  int n_edges) {
  int i = blockIdx.x * blockDim.x + threadIdx.x;
  if (i >= n_edges) return;
  atomicAdd(&acc2[dst[i]], h2pre[src[i]]);
}

// ---------------------------------------------------------------- final: out = acc2*norm_in + b2
__global__ __launch_bounds__(256) void k_final(const float* __restrict__ acc2,
                                               const float* __restrict__ norm_in,
                                               const float* __restrict__ b2,
                                               float* __restrict__ out,
                                               int n_nodes) {
  int i = blockIdx.x * blockDim.x + threadIdx.x;
  if (i >= n_nodes) return;
  out[i] = fmaf(acc2[i], norm_in[i], b2[0]);
}

extern "C" void kernel_launch(void* const* d_in, const int* in_sizes, int n_in,
                              void* d_out, int out_size, void* d_ws, size_t ws_size,
                              hipStream_t stream) {
  const float* x   = (const float*)d_in[0];
  const int*   src = (const int*)d_in[1];
  const int*   dst = (const int*)d_in[2];
  const float* W1  = (const float*)d_in[3];
  const float* b1  = (const float*)d_in[4];
  const float* W2  = (const float*)d_in[5];
  const float* b2  = (const float*)d_in[6];
  float* out = (float*)d_out;

  const int n_nodes = in_sizes[0] / N_IN;   // 100000
  const int n_edges = in_sizes[1];          // 3200000

  // Workspace layout (floats): all fits in L2 (agg/h tables are <= 3.2 MB each)
  float* ws      = (float*)d_ws;
  float* deg_out = ws;                                   // N   -> becomes norm_out
  float* deg_in  = ws + n_nodes;                         // N   -> becomes norm_in
  float* h1      = ws + 2 * (size_t)n_nodes;             // N*8
  float* agg1    = h1 + (size_t)n_nodes * N_HID;         // N*8
  float* h2pre   = agg1 + (size_t)n_nodes * N_HID;       // N
  float* acc2    = h2pre + (size_t)n_nodes;              // N
  long long zcount = 20LL * n_nodes;                     // zero everything (8 MB)

  k_zero<<<(unsigned)((zcount + 255) / 256), 256, 0, stream>>>(ws, zcount);

  k_degree<<<(n_edges + 255) / 256, 256, 0, stream>>>(src, dst, deg_out, deg_in, n_edges);

  k_norm<<<(2 * n_nodes + 255) / 256, 256, 0, stream>>>(ws, 2 * n_nodes);

  // one wave per 16-node tile, 8 waves (128 nodes) per block
  k_gemm1_wmma<<<(n_nodes + 127) / 128, 256, 0, stream>>>(x, W1, deg_out, h1, n_nodes);

  long long t1 = (long long)n_edges * N_HID;
  k_scatter1<<<(unsigned)((t1 + 255) / 256), 256, 0, stream>>>(src, dst, h1, agg1, n_edges);

  k_node_mid<<<(n_nodes + 255) / 256, 256, 0, stream>>>(agg1, deg_in, deg_out, b1, W2,
                                                        h2pre, n_nodes);

  k_scatter2<<<(n_edges + 255) / 256, 256, 0, stream>>>(src, dst, h2pre, acc2, n_edges);

  k_final<<<(n_nodes + 255) / 256, 256, 0, stream>>>(acc2, deg_in, b2, out, n_nodes);
}